// CurvatureEstimator_19894288515755
// MI455X (gfx1250) — compile-verified
//
#include <hip/hip_runtime.h>

typedef float v2f __attribute__((ext_vector_type(2)));
typedef float v8f __attribute__((ext_vector_type(8)));

#define BATCH 8
#define HS 512
#define CH 8
#define NPIX (HS * HS)                        // 262144 pixels per batch
#define DISK_AREA 709.0f                      // number of 1s in radius-15 disk
#define CURV_COEF 0.628318530717958647692f    // 3*pi/15

// Padded prefix-sum layout: per batch, rows y = -15..526 (542), cols x = -16..527 (544).
// Guard rows are zero; left pad is zero; right pad replicates the row total.
// This makes all 62 disk-conv taps unconditional (no bounds checks, no EXEC masking).
#define PROWS (HS + 30)                       // 542
#define PCOLS (HS + 32)                       // 544
#define PSTRIDE (PCOLS * CH)                  // 4352 floats per padded row

// ------------- Kernel 0: zero guard rows + Gram accumulators ---------------
__global__ __launch_bounds__(256) void k_init(float* __restrict__ Ppad,
                                              float* __restrict__ gram) {
    if (blockIdx.x == BATCH * 30) {           // last block: zero 8 x 16x16 Grams
        for (int i = threadIdx.x; i < BATCH * 256; i += 256) gram[i] = 0.0f;
        return;
    }
    const int b = blockIdx.x / 30;
    const int r = blockIdx.x % 30;
    const int prow = (r < 15) ? r : (HS + r); // 0..14 (top), 527..541 (bottom)
    float* dst = Ppad + ((size_t)b * PROWS + prow) * PSTRIDE;
    for (int t = threadIdx.x; t < PSTRIDE; t += 256) dst[t] = 0.0f;
}

// ------------- Kernel 1: per-row horizontal prefix sums (padded output) ----
// grid = BATCH*HS (one image row each), block = 256 = 8 waves (wave = channel).
__global__ __launch_bounds__(256) void k_prefix(const float* __restrict__ img,
                                                float* __restrict__ Ppad) {
    __shared__ float row[HS * CH];            // 16 KB
    const int b = blockIdx.x >> 9;
    const int y = blockIdx.x & (HS - 1);
    const size_t rbase = (size_t)blockIdx.x * (HS * CH);
    for (int t = threadIdx.x; t < HS * CH; t += 256) row[t] = img[rbase + t];
    __syncthreads();
    const int wave = threadIdx.x >> 5;        // channel
    const int lane = threadIdx.x & 31;
    const int x0 = lane * 16;
    float vals[16];
    float s = 0.f;
    #pragma unroll
    for (int k = 0; k < 16; ++k) { s += row[(x0 + k) * CH + wave]; vals[k] = s; }
    float tot = s;                            // inclusive wave scan of lane sums
    #pragma unroll
    for (int d = 1; d < 32; d <<= 1) {
        float up = __shfl_up(tot, d, 32);
        if (lane >= d) tot += up;
    }
    const float off = tot - s;
    __syncthreads();
    #pragma unroll
    for (int k = 0; k < 16; ++k) row[(x0 + k) * CH + wave] = vals[k] + off;
    __syncthreads();
    // write padded row: left pad 0, interior prefix, right pad = row total
    float* dst = Ppad + ((size_t)b * PROWS + (y + 15)) * PSTRIDE;
    for (int t = threadIdx.x; t < PSTRIDE; t += 256) {
        const int col = t >> 3;               // 0..543
        const int c   = t & 7;
        const int x   = col - 16;             // -16..527
        const int xs  = x < 0 ? 0 : (x > HS - 1 ? HS - 1 : x);
        const float v = row[xs * CH + c];
        dst[t] = (x < 0) ? 0.0f : v;
    }
}

// 3x3 max dilation: clamp-to-edge == SAME(-inf) for a max, so 9 unconditional loads
__device__ __forceinline__ float dilate3x3(const float* __restrict__ img,
                                           int b, int y, int x, int c) {
    const int ym = y > 0 ? y - 1 : 0;
    const int yp = y < HS - 1 ? y + 1 : HS - 1;
    const int xm = (x > 0 ? x - 1 : 0) * CH;
    const int xc = x * CH;
    const int xp = (x < HS - 1 ? x + 1 : HS - 1) * CH;
    const float* r0 = img + ((size_t)b * NPIX + (size_t)ym * HS) * CH + c;
    const float* r1 = img + ((size_t)b * NPIX + (size_t)y  * HS) * CH + c;
    const float* r2 = img + ((size_t)b * NPIX + (size_t)yp * HS) * CH + c;
    float m = r0[xm];
    m = fmaxf(m, r0[xc]); m = fmaxf(m, r0[xp]);
    m = fmaxf(m, r1[xm]); m = fmaxf(m, r1[xc]); m = fmaxf(m, r1[xp]);
    m = fmaxf(m, r2[xm]); m = fmaxf(m, r2[xc]); m = fmaxf(m, r2[xp]);
    return m;
}

// Disk-masked sum: 31 rows x 2 unconditional prefix lookups (padding absorbs edges)
__device__ __forceinline__ float disk_conv(const float* __restrict__ Ppad,
                                           int b, int y, int x, int c) {
    constexpr int wtab[16] = {15,14,14,14,14,14,13,13,12,12,11,10,9,7,5,0};
    // row pointer for dy = -15 is padded row (y - 15 + 15) = y
    const float* base = Ppad + ((size_t)b * PROWS + y) * PSTRIDE + c;
    float s = 0.f;
    #pragma unroll
    for (int dy = -15; dy <= 15; ++dy) {
        const int w = wtab[dy < 0 ? -dy : dy];
        const float* r = base + (size_t)(dy + 15) * PSTRIDE;
        s += r[(x + w + 16) * CH] - r[(x - w + 15) * CH];   // hi - lo, always in-range
    }
    return s;
}

// ------------- Kernel 2: fused dilate + curv + WMMA Gram accumulation ------
// Compute phase: lane = (pixel slot 0..3, channel 0..7) -> unique work, 128B
// coalesced taps. WMMA phase: 4x ds_bpermute gather the 16x4 A=W fragment
// (A layout: M = lane&15, K pair = 2*(lane>>4)); B = A^T needs identical
// per-lane data. G += W W^T via v_wmma_f32_16x16x4_f32. EXEC stays all-ones.
__global__ __launch_bounds__(256) void k_gram(const float* __restrict__ img,
                                              const float* __restrict__ Ppad,
                                              float* __restrict__ gram) {
    __shared__ float lg[8][256];
    const int b    = blockIdx.x >> 8;         // 256 blocks per batch
    const int tb   = blockIdx.x & 255;
    const int wave = threadIdx.x >> 5;
    const int lane = threadIdx.x & 31;
    // compute-phase mapping
    const int ps = lane >> 3;                 // pixel slot within quad (0..3)
    const int cc = lane & 7;                  // channel
    // wmma-phase mapping
    const bool isV = (lane & 15) >= 8;        // feature 8..15 = u*curv
    const int srcA = (lane >> 4) * 16 + (lane & 7);  // lane holding (pixel 2*(l>>4), ch)
    const int srcB = srcA + 8;                       // lane holding (pixel +1, ch)
    const int base = (tb * 8 + wave) * 128;   // 128 pixels per wave
    v8f acc = {0.f, 0.f, 0.f, 0.f, 0.f, 0.f, 0.f, 0.f};
    for (int it = 0; it < 32; ++it) {
        const int p = base + it * 4 + ps;     // quad stays within one image row
        const int y = p >> 9;
        const int x = p & (HS - 1);
        const float d    = dilate3x3(img, b, y, x, cc);
        const float conv = disk_conv(Ppad, b, y, x, cc);
        const float curv = (conv * (1.0f / DISK_AREA) - 0.5f) * CURV_COEF;
        const float dc   = d * curv;
        // gather WMMA operand: a = (W[f][k0], W[f][k0+1]) for f = lane&15
        const float d0 = __shfl(d,  srcA, 32);
        const float c0 = __shfl(dc, srcA, 32);
        const float d1 = __shfl(d,  srcB, 32);
        const float c1 = __shfl(dc, srcB, 32);
        v2f a;
        a.x = isV ? c0 : d0;
        a.y = isV ? c1 : d1;
        acc = __builtin_amdgcn_wmma_f32_16x16x4_f32(
            /*neg_a=*/false, a, /*neg_b=*/false, a,
            /*c_mod=*/(short)0, acc, /*reuse_a=*/false, /*reuse_b=*/false);
    }
    // C/D fragment: element (M, N), N = lane&15, M = reg + 8*(lane>>4)
    const int n  = lane & 15;
    const int mb = (lane >> 4) * 8;
    #pragma unroll
    for (int r = 0; r < 8; ++r) lg[wave][(mb + r) * 16 + n] = acc[r];
    __syncthreads();
    {
        const int t = threadIdx.x;            // one Gram element per thread
        float s = 0.f;
        #pragma unroll
        for (int w = 0; w < 8; ++w) s += lg[w][t];
        atomicAdd(&gram[b * 256 + t], s);
    }
}

// ------------- Kernel 3: [8, 28] epilogue from Gram matrices ---------------
__global__ void k_final(const float* __restrict__ gram, float* __restrict__ out) {
    const int t = threadIdx.x;
    if (t >= BATCH * 28) return;
    const int b = t / 28;
    int p = t % 28;
    int i = 0, j = 1;                         // decode pair (i<j, row-major order)
    for (int a = 0; a < CH; ++a) {
        const int rem = CH - 1 - a;
        if (p < rem) { i = a; j = a + 1 + p; break; }
        p -= rem;
    }
    const float* G = gram + b * 256;
    const float num = 0.5f * (G[(8 + i) * 16 + j] - G[(8 + j) * 16 + i]);
    const float den = G[i * 16 + j];
    out[t] = num / den;
}

extern "C" void kernel_launch(void* const* d_in, const int* in_sizes, int n_in,
                              void* d_out, int out_size, void* d_ws, size_t ws_size,
                              hipStream_t stream) {
    const float* img = (const float*)d_in[0];        // [8, 512, 512, 8] f32
    float* out  = (float*)d_out;                     // [8, 28] f32
    float* Ppad = (float*)d_ws;                      // 8*542*544*8 f32 = 75.5 MB
    float* gram = Ppad + (size_t)BATCH * PROWS * PSTRIDE;  // 8 * 256 f32

    k_init  <<<BATCH * 30 + 1, 256, 0, stream>>>(Ppad, gram);
    k_prefix<<<BATCH * HS,     256, 0, stream>>>(img, Ppad);
    k_gram  <<<BATCH * 256,    256, 0, stream>>>(img, Ppad, gram);
    k_final <<<1,              256, 0, stream>>>(gram, out);
}